// CrossViewConstraintsLoss_48773648613475
// MI455X (gfx1250) — compile-verified
//
#include <hip/hip_runtime.h>
#include <hip/hip_bf16.h>
#include <stdint.h>

// CDNA5 (gfx1250) wave32 WMMA vector types
typedef __attribute__((ext_vector_type(2))) float v2f;   // A/B frag of V_WMMA_F32_16X16X4_F32
typedef __attribute__((ext_vector_type(8))) float v8f;   // 16x16 f32 C/D frag

#define N_PTS   4096
#define DIMS    512
#define GTHR    0.05f
#define PD_EPS  1e-6f
#define NORM_EPSF 1e-12f

// ---------------------------------------------------------------- utilities
__device__ __forceinline__ float urand_hash(uint32_t v, uint32_t i, uint32_t j) {
    uint32_t x = (v + 1u) * 0x9E3779B9u ^ (i * 0x85EBCA6Bu) ^ (j * 0xC2B2AE35u);
    x ^= x >> 16; x *= 0x7FEB352Du;
    x ^= x >> 15; x *= 0x846CA68Bu;
    x ^= x >> 16;
    return (float)(x >> 8) * (1.0f / 16777216.0f);   // [0,1)
}

// async global->LDS 16B per lane (gfx1250, tracked by ASYNCcnt)
__device__ __forceinline__ void async_copy_b128(uint32_t lds_off, uint64_t gaddr) {
    asm volatile("global_load_async_to_lds_b128 %0, %1, off"
                 :: "v"(lds_off), "v"(gaddr) : "memory");
}

// ---------------------------------------------------------------- init
__global__ void init_ws_kernel(unsigned int* __restrict__ mask32,
                               float* __restrict__ lossSum,
                               unsigned int* __restrict__ posCount) {
    size_t idx = (size_t)blockIdx.x * blockDim.x + threadIdx.x;
    if (idx < (size_t)N_PTS * N_PTS / 4) mask32[idx] = 0u;
    if (idx < 3) lossSum[idx] = 0.0f;
    if (idx == 3) *posCount = 0u;
}

// ---------------------------------------------------------------- row L2 normalize
__global__ void normalize_rows_kernel(const float* __restrict__ X,
                                      float* __restrict__ Xn,
                                      float* __restrict__ rs,
                                      float* __restrict__ s2) {
    __shared__ float red[256];
    const int row = blockIdx.x;
    const int tid = threadIdx.x;
    const float* xr = X + (size_t)row * DIMS;
    float a0 = xr[tid];
    float a1 = xr[tid + 256];
    red[tid] = a0 * a0 + a1 * a1;
    __syncthreads();
    for (int s = 128; s > 0; s >>= 1) {
        if (tid < s) red[tid] += red[tid + s];
        __syncthreads();
    }
    const float sumsq = red[0];
    __syncthreads();
    const float inv = 1.0f / fmaxf(sqrtf(sumsq), NORM_EPSF);
    const float n0 = a0 * inv, n1 = a1 * inv;
    Xn[(size_t)row * DIMS + tid]       = n0;
    Xn[(size_t)row * DIMS + tid + 256] = n1;
    red[tid] = n0 + n1;
    __syncthreads();
    for (int s = 128; s > 0; s >>= 1) {
        if (tid < s) red[tid] += red[tid + s];
        __syncthreads();
    }
    if (tid == 0) {
        rs[row] = red[0];
        s2[row] = sumsq * inv * inv;   // == sum(Xn_row^2)
    }
}

// ---------------------------------------------------------------- Gram matrix via f32 WMMA
// G = Xn * Xn^T.  256 threads = 8 waves; 64x64 C tile per block.
// Double-buffered 64x32 K panels in LDS, filled by async global->LDS B128
// (4 async instructions per wave per panel -> s_wait_asynccnt 4 overlaps
//  next-panel fetch with current-panel WMMA compute).
__global__ void __launch_bounds__(256) gram_wmma_kernel(const float* __restrict__ Xn,
                                                        float* __restrict__ G) {
    __shared__ __align__(16) float As[2][64][36];   // [buf][m][k], padded stride (144B rows)
    __shared__ __align__(16) float Bs[2][64][36];   // [buf][n][k]

    const int tid  = threadIdx.x;
    const int lane = tid & 31;
    const int gm0 = blockIdx.y * 64;
    const int gn0 = blockIdx.x * 64;

    const int t0  = (tid >> 5) * 2;   // tiles t0, t0+1 of the 4x4 tile grid
    const int tm  = t0 >> 2;          // shared row-tile
    const int tn0 = t0 & 3;           // col-tiles tn0, tn0+1

    v8f acc0 = {};
    v8f acc1 = {};

    const int mrow  = lane & 15;          // M (or N) index inside 16x16 tile
    const int khalf = (lane >> 4) << 1;   // lanes 16-31 hold K+2,K+3 (ISA 7.12.2)

    // per-thread staging coordinates: 2 x B128 per matrix per panel
    const int r0 = tid >> 2;               // 0..63
    const int c0 = (tid & 3) << 3;         // 0,8,16,24  (two float4 slots: c0, c0+4)

    auto issue_panel = [&](int kb, int b) {
        #pragma unroll
        for (int h = 0; h < 2; ++h) {
            const int c4 = c0 + h * 4;
            async_copy_b128((uint32_t)(uintptr_t)&As[b][r0][c4],
                            (uint64_t)(uintptr_t)(Xn + (size_t)(gm0 + r0) * DIMS + kb + c4));
            async_copy_b128((uint32_t)(uintptr_t)&Bs[b][r0][c4],
                            (uint64_t)(uintptr_t)(Xn + (size_t)(gn0 + r0) * DIMS + kb + c4));
        }
    };

    int buf = 0;
    issue_panel(0, 0);

    for (int kb = 0; kb < DIMS; kb += 32) {
        const bool has_next = (kb + 32 < DIMS);
        if (has_next) issue_panel(kb + 32, buf ^ 1);

        // async loads complete in order: <=4 outstanding means current panel landed
        if (has_next) asm volatile("s_wait_asynccnt 0x4" ::: "memory");
        else          asm volatile("s_wait_asynccnt 0x0" ::: "memory");
        __syncthreads();                 // all waves' panel writes visible

        #pragma unroll
        for (int kk = 0; kk < 32; kk += 4) {
            const int k0 = kk + khalf;
            const v2f a  = *(const v2f*)&As[buf][tm  * 16 + mrow][k0];
            const v2f b0 = *(const v2f*)&Bs[buf][tn0 * 16 + mrow][k0];
            const v2f b1 = *(const v2f*)&Bs[buf][(tn0 + 1) * 16 + mrow][k0];
            acc0 = __builtin_amdgcn_wmma_f32_16x16x4_f32(false, a, false, b0,
                                                         (short)0, acc0, false, false);
            acc1 = __builtin_amdgcn_wmma_f32_16x16x4_f32(false, a, false, b1,
                                                         (short)0, acc1, false, false);
        }
        __syncthreads();                 // buffer reusable for panel kb+64's issue
        buf ^= 1;
    }

    // C/D layout: VGPR r holds M=r (lanes 0-15) and M=8+r (lanes 16-31), N = lane&15
    const int nc  = lane & 15;
    const int mhi = (lane >> 4) << 3;
    #pragma unroll
    for (int r = 0; r < 8; ++r) {
        const int row = gm0 + tm * 16 + mhi + r;
        G[(size_t)row * N_PTS + gn0 + tn0 * 16 + nc]       = acc0[r];
        G[(size_t)row * N_PTS + gn0 + (tn0 + 1) * 16 + nc] = acc1[r];
    }
}

// ---------------------------------------------------------------- union graph threshold
__global__ void union_thresh_kernel(const float* __restrict__ G,
                                    unsigned char* __restrict__ mask) {
    const size_t idx = (size_t)blockIdx.x * blockDim.x + threadIdx.x;
    if (G[idx] > GTHR) mask[idx] = 1;
}

// ---------------------------------------------------------------- per-row inclusive scan of zero-mask
__global__ void row_scan_kernel(const unsigned char* __restrict__ mask,
                                unsigned short* __restrict__ csum,
                                int* __restrict__ cnt,
                                unsigned int* __restrict__ posCount) {
    __shared__ int sc[256];
    __shared__ int carry;
    const int i = blockIdx.x;
    const int tid = threadIdx.x;
    if (tid == 0) carry = 0;
    __syncthreads();
    for (int c = 0; c < 16; ++c) {
        const int j = c * 256 + tid;
        const int z = (i == j) ? 1 : ((mask[(size_t)i * N_PTS + j] == 0) ? 1 : 0);
        sc[tid] = z;
        __syncthreads();
        for (int off = 1; off < 256; off <<= 1) {    // Hillis-Steele inclusive scan
            const int v = (tid >= off) ? sc[tid - off] : 0;
            __syncthreads();
            sc[tid] += v;
            __syncthreads();
        }
        const int incl = sc[tid];
        csum[(size_t)i * N_PTS + j] = (unsigned short)(carry + incl);
        __syncthreads();
        if (tid == 255) carry += incl;
        __syncthreads();
    }
    if (tid == 255) {
        cnt[i] = carry;
        atomicAdd(posCount, (unsigned int)(N_PTS - carry));
    }
}

// ---------------------------------------------------------------- per-view triplet hinge loss
__global__ void loss_view_kernel(const float* __restrict__ G,
                                 const unsigned char* __restrict__ mask,
                                 const unsigned short* __restrict__ csum,
                                 const int* __restrict__ cnt,
                                 const float* __restrict__ rs,
                                 const float* __restrict__ s2,
                                 float* __restrict__ lossSum,
                                 int view) {
    __shared__ float red[256];
    const int i = blockIdx.x;
    const int tid = threadIdx.x;
    const float s2i = s2[i];
    const float rsi = rs[i];
    const int ci = cnt[i];
    const unsigned short* cs = csum + (size_t)i * N_PTS;
    const float* gr = G + (size_t)i * N_PTS;
    const float dterm = (float)DIMS * PD_EPS * PD_EPS;
    float acc = 0.0f;
    for (int c = 0; c < 16; ++c) {
        const int j = c * 256 + tid;
        const bool pos = (i != j) && (mask[(size_t)i * N_PTS + j] != 0);
        if (pos) {
            const float u = urand_hash((uint32_t)view, (uint32_t)i, (uint32_t)j);
            int r = (int)(u * (float)ci);
            r = min(r, max(ci - 1, 0));
            // searchsorted(csum_row, r, side='right'): first idx with csum > r
            int lo = 0, hi = N_PTS;
            while (lo < hi) {
                const int mid = (lo + hi) >> 1;
                if ((int)cs[mid] <= r) lo = mid + 1; else hi = mid;
            }
            const int nidx = lo;
            const float gij = gr[j];
            const float gin = gr[nidx];
            const float d_ij = sqrtf(fmaxf(
                s2i + s2[j] - 2.0f * gij + 2.0f * PD_EPS * (rsi - rs[j]) + dterm, 0.0f));
            const float d_an = sqrtf(fmaxf(
                s2i + s2[nidx] - 2.0f * gin + 2.0f * PD_EPS * (rsi - rs[nidx]) + dterm, 0.0f));
            acc += fmaxf(d_ij - d_an + 1.0f, 0.0f);
        }
    }
    red[tid] = acc;
    __syncthreads();
    for (int s = 128; s > 0; s >>= 1) {
        if (tid < s) red[tid] += red[tid + s];
        __syncthreads();
    }
    if (tid == 0) atomicAdd(&lossSum[view], red[0]);
}

// ---------------------------------------------------------------- finalize
__global__ void finalize_kernel(const float* __restrict__ lossSum,
                                const unsigned int* __restrict__ posCount,
                                float* __restrict__ out) {
    if (threadIdx.x == 0 && blockIdx.x == 0) {
        const float M = (float)(*posCount);
        out[0] = (lossSum[0] + lossSum[1] + lossSum[2]) / M;
    }
}

// ---------------------------------------------------------------- launch
extern "C" void kernel_launch(void* const* d_in, const int* in_sizes, int n_in,
                              void* d_out, int out_size, void* d_ws, size_t ws_size,
                              hipStream_t stream) {
    (void)in_sizes; (void)n_in; (void)out_size; (void)ws_size;
    const float* X[3] = { (const float*)d_in[0], (const float*)d_in[1], (const float*)d_in[2] };

    // workspace layout (descending alignment): f32 blocks, then u16, then u8
    char* ws = (char*)d_ws;
    size_t off = 0;
    float* Xn[3];
    for (int v = 0; v < 3; ++v) { Xn[v] = (float*)(ws + off); off += (size_t)N_PTS * DIMS * sizeof(float); }
    float* G = (float*)(ws + off);             off += (size_t)N_PTS * N_PTS * sizeof(float);
    float* rs[3]; float* s2[3];
    for (int v = 0; v < 3; ++v) { rs[v] = (float*)(ws + off); off += N_PTS * sizeof(float); }
    for (int v = 0; v < 3; ++v) { s2[v] = (float*)(ws + off); off += N_PTS * sizeof(float); }
    int* cnt = (int*)(ws + off);               off += N_PTS * sizeof(int);
    float* lossSum = (float*)(ws + off);       off += 4 * sizeof(float);
    unsigned int* posCount = (unsigned int*)(ws + off); off += sizeof(unsigned int) * 4;
    unsigned short* csum = (unsigned short*)(ws + off); off += (size_t)N_PTS * N_PTS * sizeof(unsigned short);
    unsigned char* mask = (unsigned char*)(ws + off);   off += (size_t)N_PTS * N_PTS;

    // zero the union mask + accumulators
    init_ws_kernel<<<(N_PTS * (size_t)N_PTS / 4 + 255) / 256, 256, 0, stream>>>(
        (unsigned int*)mask, lossSum, posCount);

    // normalize all views
    for (int v = 0; v < 3; ++v)
        normalize_rows_kernel<<<N_PTS, 256, 0, stream>>>(X[v], Xn[v], rs[v], s2[v]);

    const dim3 gemm_grid(N_PTS / 64, N_PTS / 64);

    // pass 1: union graph (G buffer reused per view)
    for (int v = 0; v < 3; ++v) {
        gram_wmma_kernel<<<gemm_grid, 256, 0, stream>>>(Xn[v], G);
        union_thresh_kernel<<<(N_PTS * (size_t)N_PTS) / 256, 256, 0, stream>>>(G, mask);
    }

    // row prefix sums of zero-mask + positive count
    row_scan_kernel<<<N_PTS, 256, 0, stream>>>(mask, csum, cnt, posCount);

    // pass 2: per-view loss (recompute Gram, derive distances from it)
    for (int v = 0; v < 3; ++v) {
        gram_wmma_kernel<<<gemm_grid, 256, 0, stream>>>(Xn[v], G);
        loss_view_kernel<<<N_PTS, 256, 0, stream>>>(G, mask, csum, cnt, rs[v], s2[v], lossSum, v);
    }

    finalize_kernel<<<1, 1, 0, stream>>>(lossSum, posCount, (float*)d_out);
}